// RoIAlign_3d_23252952941136
// MI455X (gfx1250) — compile-verified
//
#include <hip/hip_runtime.h>
#include <math.h>

// Problem constants (fixed by the reference).
constexpr int C  = 128, D = 32, H = 64, W = 64;
constexpr int PD = 4, PH = 7, PW = 7, SR = 2;
constexpr int BINS = PD * PH * PW;          // 196
constexpr float SCALE = 0.25f;
constexpr int HW  = H * W;
constexpr int DHW = D * H * W;

#define USE_ASYNC_LDS 1

// ---------------------------------------------------------------------------
// Kernel 1: NCDHW -> NDHWC transpose of the feature tensor.
// One block per (n, d, h) slice: [C=128][W=64] -> [W=64][C=128].
// Loads are staged into LDS with CDNA5 async global->LDS copies
// (global_load_async_to_lds_b32, tracked by ASYNCcnt), then written out
// coalesced along C with b128 stores.
// ---------------------------------------------------------------------------
__global__ __launch_bounds__(256) void transpose_ndhwc(
    const float* __restrict__ src, float* __restrict__ dst, int slices) {
  __shared__ float tile[W * (C + 1)];   // [w][c] rows padded to 129 floats
  int slice = blockIdx.x;               // n*D*H + d*H + h
  if (slice >= slices) return;
  int tid = threadIdx.x;
  int n   = slice / (D * H);
  int dh  = slice % (D * H);
  const float* s = src + (size_t)n * C * DHW + (size_t)dh * W;

  // Load phase: 32 elements per thread, coalesced along w.
#pragma unroll 4
  for (int i = 0; i < (C * W) / 256; ++i) {
    int idx = tid + i * 256;
    int c = idx / W;
    int w = idx % W;
    const float* g = s + (size_t)c * DHW + w;
#if USE_ASYNC_LDS
    // ptrtoint of the element address: escapes `tile` (so the compiler must
    // honor the memory clobber and keep the later ds_loads), and its low 32
    // bits are exactly the wave-relative LDS byte offset the async VDST wants
    // (ISA 10.2: LDS flat addr low 32 bits == LDS offset).
    unsigned lofs = (unsigned)(uintptr_t)(&tile[w * (C + 1) + c]);
    asm volatile("global_load_async_to_lds_b32 %0, %1, off"
                 :: "v"(lofs), "v"(g) : "memory");
#else
    tile[w * (C + 1) + c] = *g;
#endif
  }
#if USE_ASYNC_LDS
  asm volatile("s_wait_asynccnt 0" ::: "memory");
#endif
  __syncthreads();

  // Store phase: gather 4 consecutive channels from LDS, store one float4.
  // Consecutive lanes -> consecutive c4 -> 512B coalesced wave stores.
  float4* dsl4 = (float4*)(dst + (size_t)slice * (size_t)(W * C));
#pragma unroll
  for (int i = 0; i < (C * W) / (256 * 4); ++i) {
    int idx4 = tid + i * 256;                // float4 index within slice
    int w  = idx4 / (C / 4);
    int c4 = (idx4 % (C / 4)) * 4;
    const float* trow = &tile[w * (C + 1) + c4];
    float4 v;
    v.x = trow[0]; v.y = trow[1]; v.z = trow[2]; v.w = trow[3];
    dsl4[idx4] = v;
  }
}

// ---------------------------------------------------------------------------
// Kernel 2: RoI Align gather. One wave32 per (roi, bin); lane covers 4
// channels. CHLAST=true reads NDHWC (float4 per lane, 512B coalesced wave
// loads per corner); CHLAST=false is the NCDHW fallback.
// Output layout: (R, C, PD, PH, PW).
// ---------------------------------------------------------------------------
template <bool CHLAST>
__global__ __launch_bounds__(256) void roi_gather(
    const float* __restrict__ feat, const float* __restrict__ rois,
    float* __restrict__ out, int R) {
  int wave = threadIdx.x >> 5;
  int lane = threadIdx.x & 31;
  int gb = blockIdx.x * 8 + wave;
  if (gb >= R * BINS) return;
  int r   = gb / BINS;
  int bin = gb - r * BINS;
  int pd  = bin / (PH * PW);
  int rem = bin - pd * (PH * PW);
  int ph  = rem / PW;
  int pw  = rem - ph * PW;

  const float* roi = rois + (size_t)r * 7;
  int   b  = (int)roi[0];
  float x1 = roi[1] * SCALE, y1 = roi[2] * SCALE, z1 = roi[3] * SCALE;
  float x2 = roi[4] * SCALE, y2 = roi[5] * SCALE, z2 = roi[6] * SCALE;
  float rd = fmaxf(z2 - z1, 1.0f);
  float rh = fmaxf(y2 - y1, 1.0f);
  float rw = fmaxf(x2 - x1, 1.0f);
  float bz = rd / (float)PD, by = rh / (float)PH, bx = rw / (float)PW;

  const float* fb = feat + (size_t)b * DHW * C;   // same offset either layout
  int c0 = lane * 4;

  float a0 = 0.f, a1 = 0.f, a2 = 0.f, a3 = 0.f;

#pragma unroll
  for (int iz = 0; iz < SR; ++iz) {
    float z = z1 + ((float)(pd * SR + iz) + 0.5f) * (bz / (float)SR);
    if (z < -1.0f || z > (float)D) continue;
    float cz  = fminf(fmaxf(z, 0.0f), (float)(D - 1));
    int   zlo = (int)cz;
    int   zhi = (zlo + 1 < D) ? zlo + 1 : D - 1;
    float wz  = cz - (float)zlo;
#pragma unroll
    for (int iy = 0; iy < SR; ++iy) {
      float y = y1 + ((float)(ph * SR + iy) + 0.5f) * (by / (float)SR);
      if (y < -1.0f || y > (float)H) continue;
      float cy  = fminf(fmaxf(y, 0.0f), (float)(H - 1));
      int   ylo = (int)cy;
      int   yhi = (ylo + 1 < H) ? ylo + 1 : H - 1;
      float wy  = cy - (float)ylo;
#pragma unroll
      for (int ix = 0; ix < SR; ++ix) {
        float x = x1 + ((float)(pw * SR + ix) + 0.5f) * (bx / (float)SR);
        if (x < -1.0f || x > (float)W) continue;
        float cx  = fminf(fmaxf(x, 0.0f), (float)(W - 1));
        int   xlo = (int)cx;
        int   xhi = (xlo + 1 < W) ? xlo + 1 : W - 1;
        float wx  = cx - (float)xlo;

        float w00 = (1.f - wz) * (1.f - wy);
        float w01 = (1.f - wz) * wy;
        float w10 = wz * (1.f - wy);
        float w11 = wz * wy;
        float wx0 = 1.f - wx;

        if (CHLAST) {
          // Prefetch the zhi planes while the zlo corner loads are issued.
          __builtin_prefetch(fb + ((size_t)(zhi * H + ylo) * W + xlo) * C + c0, 0, 0);
          __builtin_prefetch(fb + ((size_t)(zhi * H + yhi) * W + xlo) * C + c0, 0, 0);
          const float4* p000 = (const float4*)(fb + ((size_t)(zlo * H + ylo) * W + xlo) * C) + lane;
          const float4* p001 = (const float4*)(fb + ((size_t)(zlo * H + ylo) * W + xhi) * C) + lane;
          const float4* p010 = (const float4*)(fb + ((size_t)(zlo * H + yhi) * W + xlo) * C) + lane;
          const float4* p011 = (const float4*)(fb + ((size_t)(zlo * H + yhi) * W + xhi) * C) + lane;
          const float4* p100 = (const float4*)(fb + ((size_t)(zhi * H + ylo) * W + xlo) * C) + lane;
          const float4* p101 = (const float4*)(fb + ((size_t)(zhi * H + ylo) * W + xhi) * C) + lane;
          const float4* p110 = (const float4*)(fb + ((size_t)(zhi * H + yhi) * W + xlo) * C) + lane;
          const float4* p111 = (const float4*)(fb + ((size_t)(zhi * H + yhi) * W + xhi) * C) + lane;
          float4 v000 = *p000, v001 = *p001, v010 = *p010, v011 = *p011;
          float4 v100 = *p100, v101 = *p101, v110 = *p110, v111 = *p111;
#define TRI_ACC(v, wgt)                                                      \
          a0 = fmaf((v).x, (wgt), a0); a1 = fmaf((v).y, (wgt), a1);          \
          a2 = fmaf((v).z, (wgt), a2); a3 = fmaf((v).w, (wgt), a3);
          TRI_ACC(v000, w00 * wx0); TRI_ACC(v001, w00 * wx);
          TRI_ACC(v010, w01 * wx0); TRI_ACC(v011, w01 * wx);
          TRI_ACC(v100, w10 * wx0); TRI_ACC(v101, w10 * wx);
          TRI_ACC(v110, w11 * wx0); TRI_ACC(v111, w11 * wx);
        } else {
          // NCDHW fallback: scalar loads, channel stride = DHW.
          size_t s00 = (size_t)zlo * HW + (size_t)ylo * W;
          size_t s01 = (size_t)zlo * HW + (size_t)yhi * W;
          size_t s10 = (size_t)zhi * HW + (size_t)ylo * W;
          size_t s11 = (size_t)zhi * HW + (size_t)yhi * W;
#pragma unroll
          for (int k = 0; k < 4; ++k) {
            const float* fc = fb + (size_t)(c0 + k) * DHW;
            float v = fc[s00 + xlo] * (w00 * wx0) + fc[s00 + xhi] * (w00 * wx)
                    + fc[s01 + xlo] * (w01 * wx0) + fc[s01 + xhi] * (w01 * wx)
                    + fc[s10 + xlo] * (w10 * wx0) + fc[s10 + xhi] * (w10 * wx)
                    + fc[s11 + xlo] * (w11 * wx0) + fc[s11 + xhi] * (w11 * wx);
            if (k == 0) a0 += v; else if (k == 1) a1 += v;
            else if (k == 2) a2 += v; else a3 += v;
          }
        }
      }
    }
  }

  const float inv = 1.0f / (float)(SR * SR * SR);
  size_t ob = ((size_t)r * C + c0) * BINS + bin;   // (R, C, PD, PH, PW)
  out[ob]            = a0 * inv;
  out[ob + BINS]     = a1 * inv;
  out[ob + 2 * BINS] = a2 * inv;
  out[ob + 3 * BINS] = a3 * inv;
}

// ---------------------------------------------------------------------------
extern "C" void kernel_launch(void* const* d_in, const int* in_sizes, int n_in,
                              void* d_out, int out_size, void* d_ws, size_t ws_size,
                              hipStream_t stream) {
  const float* features = (const float*)d_in[0];
  const float* rois     = (const float*)d_in[1];
  float* out = (float*)d_out;

  int R = in_sizes[1] / 7;
  int N = in_sizes[0] / (C * DHW);
  int total_bins = R * BINS;
  int gather_blocks = (total_bins + 7) / 8;   // 8 wave32 bins per 256-thr block

  size_t need = (size_t)in_sizes[0] * sizeof(float);
  if (ws_size >= need) {
    // Fast path: one-time NCDHW -> NDHWC transpose into scratch (fits in the
    // 192MB L2), then fully coalesced channels-last gather.
    float* ft = (float*)d_ws;
    int slices = N * D * H;
    transpose_ndhwc<<<slices, 256, 0, stream>>>(features, ft, slices);
    roi_gather<true><<<gather_blocks, 256, 0, stream>>>(ft, rois, out, R);
  } else {
    roi_gather<false><<<gather_blocks, 256, 0, stream>>>(features, rois, out, R);
  }
}